// MambaBlock_35871566856359
// MI455X (gfx1250) — compile-verified
//
#include <hip/hip_runtime.h>
#include <math.h>

// ---- problem constants (match reference) ----
#define D_MODEL    768
#define D_STATE    64
#define D_CONVW    4
#define HEADDIM    64
#define D_INNER    1536
#define NHEADS     24
#define CONV_DIM   1664          // D_INNER + 2*D_STATE
#define D_IN_PROJ  3224          // 2*D_INNER + 2*D_STATE + NHEADS
#define SEQ_B      2
#define SEQ_L      4096
#define ROWS       (SEQ_B * SEQ_L)   // 8192
#define NCHUNK     (SEQ_L / 64)      // 64
#define EPSV       1e-5f

typedef __attribute__((ext_vector_type(2))) float v2f;
typedef __attribute__((ext_vector_type(8))) float v8f;

static __device__ __forceinline__ v8f wmma4(v2f a, v2f b, v8f c) {
  // V_WMMA_F32_16X16X4_F32 : D = A(16x4) * B(4x16) + C(16x16), full f32
  return __builtin_amdgcn_wmma_f32_16x16x4_f32(false, a, false, b, (short)0, c,
                                               false, false);
}

static __device__ __forceinline__ float sigmoidf_(float v) {
  return 1.0f / (1.0f + __expf(-v));
}

// ---- optional gfx1250 async global->LDS path (ASYNCcnt engines) ----
#if defined(__has_builtin)
#if __has_builtin(__builtin_amdgcn_global_load_async_to_lds_b128) && \
    __has_builtin(__builtin_amdgcn_s_wait_asynccnt)
#define HAVE_ASYNC_LDS 1
#endif
#endif

#ifdef HAVE_ASYNC_LDS
// exact pointee type per hipcc diagnostic: 'vector_size(4*sizeof(int)) int'
typedef int v4i_ __attribute__((vector_size(4 * sizeof(int))));
typedef __attribute__((address_space(1))) v4i_ g_v4i;   // global (__device__)
typedef __attribute__((address_space(3))) v4i_ l_v4i;   // LDS
static __device__ __forceinline__ void async_cp16(const float* g, float* l) {
  // generic->AS casts via integers: low 32 bits of a generic LDS pointer are
  // the LDS byte offset (ISA 10.2 aperture rules); global generic == AS1.
  __builtin_amdgcn_global_load_async_to_lds_b128(
      (g_v4i*)(unsigned long long)(uintptr_t)g,
      (l_v4i*)(unsigned)(uintptr_t)l, 0, 0);
}
static __device__ __forceinline__ void async_wait0() {
  __builtin_amdgcn_s_wait_asynccnt(0);
}
#endif

// =====================================================================
// Kernel: generic f32 WMMA GEMM  C[M,N] = A[M,K] * B[K,N]
// block = 256 threads (8 waves); block tile 64 rows x 128 cols;
// wave tile 16 rows x 64 cols (4 accumulators); K staged in LDS, step 32.
// Requires: M % 64 == 0, K % 32 == 0, N % 4 == 0. N tail masked.
// =====================================================================
#define KSTEP 32
__global__ __launch_bounds__(256) void gemm_wmma_f32(
    const float* __restrict__ A, const float* __restrict__ B,
    float* __restrict__ C, int M, int N, int K) {
  __shared__ float sA[64 * KSTEP];   // 64 rows x 32 k
  __shared__ float sB[KSTEP * 128];  // 32 k  x 128 cols

  const int tid  = threadIdx.x;
  const int wave = tid >> 5;
  const int lane = tid & 31;
  const int hl   = lane >> 4;
  const int l15  = lane & 15;

  const int rowBlk = blockIdx.y * 64;
  const int colBlk = blockIdx.x * 128;
  const int wr = (wave >> 1) * 16;
  const int wc = (wave & 1) * 64;
  const bool fullTile = (colBlk + 128) <= N;

  v8f acc[4] = {};

  for (int k0 = 0; k0 < K; k0 += KSTEP) {
    __syncthreads();
#ifdef HAVE_ASYNC_LDS
    // A slice 64x32 : 512 x b128 async copies
    for (int i = tid; i < 512; i += 256) {
      int r = i >> 3, c4 = (i & 7) * 4;
      async_cp16(&A[(size_t)(rowBlk + r) * K + (k0 + c4)], &sA[r * KSTEP + c4]);
    }
    if (fullTile) {
      // B slice 32x128 : 1024 x b128 async copies
      for (int i = tid; i < 1024; i += 256) {
        int r = i >> 5, c4 = (i & 31) * 4;
        async_cp16(&B[(size_t)(k0 + r) * N + colBlk + c4], &sB[r * 128 + c4]);
      }
    } else {
      for (int i = tid; i < KSTEP * 128; i += 256) {
        int r = i >> 7, c = i & 127;
        int n = colBlk + c;
        sB[i] = (n < N) ? B[(size_t)(k0 + r) * N + n] : 0.0f;
      }
    }
    async_wait0();
#else
    for (int i = tid; i < 64 * KSTEP; i += 256) {
      int r = i >> 5, c = i & 31;
      sA[i] = A[(size_t)(rowBlk + r) * K + (k0 + c)];
    }
    for (int i = tid; i < KSTEP * 128; i += 256) {
      int r = i >> 7, c = i & 127;
      int n = colBlk + c;
      sB[i] = (n < N) ? B[(size_t)(k0 + r) * N + n] : 0.0f;
    }
#endif
    if (k0 + KSTEP < K && tid < 64) {
      __builtin_prefetch(&A[(size_t)(rowBlk + tid) * K + (k0 + KSTEP)], 0, 1);
    }
    __syncthreads();

    for (int kk = 0; kk < KSTEP; kk += 4) {
      const int ka = kk + hl * 2;
      v2f a;
      a.x = sA[(wr + l15) * KSTEP + ka];
      a.y = sA[(wr + l15) * KSTEP + ka + 1];
#pragma unroll
      for (int j = 0; j < 4; ++j) {
        const int col = wc + j * 16 + l15;
        v2f b;
        b.x = sB[ka * 128 + col];
        b.y = sB[(ka + 1) * 128 + col];
        acc[j] = wmma4(a, b, acc[j]);
      }
    }
  }

#pragma unroll
  for (int j = 0; j < 4; ++j) {
#pragma unroll
    for (int i = 0; i < 8; ++i) {
      int r = rowBlk + wr + i + hl * 8;
      int c = colBlk + wc + j * 16 + l15;
      if (c < N) C[(size_t)r * N + c] = acc[j][i];
    }
  }
}

// =====================================================================
// Kernel: dt softplus + dA = exp(dt * A)
// =====================================================================
__global__ __launch_bounds__(256) void prep_dt(
    const float* __restrict__ zx, const float* __restrict__ dt_bias,
    const float* __restrict__ A_log, float* __restrict__ dtsp,
    float* __restrict__ dAv, int total) {
  int i = blockIdx.x * 256 + threadIdx.x;
  if (i >= total) return;
  int h = i % NHEADS;
  size_t row = (size_t)i / NHEADS;
  float v = zx[row * D_IN_PROJ + (D_INNER + CONV_DIM) + h] + dt_bias[h];
  float sp = (v > 20.0f) ? v : log1pf(__expf(v));
  float a = -__expf(A_log[h]);
  dtsp[i] = sp;
  dAv[i]  = __expf(sp * a);
}

// =====================================================================
// Kernel: causal depthwise conv (width 4) + bias + silu
// =====================================================================
__global__ __launch_bounds__(256) void conv_silu(
    const float* __restrict__ zx, const float* __restrict__ cw,
    const float* __restrict__ cb, float* __restrict__ out) {
  size_t i = (size_t)blockIdx.x * 256 + threadIdx.x;
  size_t total = (size_t)ROWS * CONV_DIM;
  if (i >= total) return;
  int c = (int)(i % CONV_DIM);
  size_t bl = i / CONV_DIM;
  int l = (int)(bl % SEQ_L);
  float acc = cb[c];
#pragma unroll
  for (int k = 0; k < D_CONVW; ++k) {
    int lp = l - (D_CONVW - 1) + k;
    if (lp >= 0) {
      acc += zx[(bl + (size_t)(k - (D_CONVW - 1))) * D_IN_PROJ + D_INNER + c] *
             cw[c * D_CONVW + k];
    }
  }
  out[i] = acc * sigmoidf_(acc);
}

// =====================================================================
// SSD scan, three passes. LC=64 chunks; LDS stride 68: row starts 16B
// aligned (async b128 legal), column walks hit banks 4l mod 64 (distinct),
// all fragment pairs 8B aligned -> ds_load_b64.
// =====================================================================
#define LC 64
#define ST 68

// ---- Pass A: per-(b,h,chunk) local state  S_c = cumL * sum_s icum_s u_s B_s^T
__global__ __launch_bounds__(128) void scan_stateA(
    const float* __restrict__ xbcc, const float* __restrict__ dtsp,
    const float* __restrict__ dAv, float* __restrict__ Sloc,
    float* __restrict__ cumL) {
  __shared__ float sB[LC * ST];
  __shared__ float sU[LC * ST];
  __shared__ float sa[LC], scum[LC], sic[LC];

  const int bh = blockIdx.x, cidx = blockIdx.y;
  const int b = bh / NHEADS, h = bh % NHEADS;
  const int tid = threadIdx.x;
  const int wave = tid >> 5, lane = tid & 31, hl = lane >> 4, l15 = lane & 15;
  const int wrow = wave * 16;
  const size_t rowbase = (size_t)b * SEQ_L + cidx * LC;

#ifdef HAVE_ASYNC_LDS
  for (int i = tid; i < LC * 16; i += 128) {  // B rows: 64 x 16 b128 chunks
    int t = i >> 4, c4 = (i & 15) * 4;
    async_cp16(&xbcc[(rowbase + t) * CONV_DIM + D_INNER + c4], &sB[t * ST + c4]);
  }
#endif
  for (int i = tid; i < LC * 64; i += 128) {
    int t = i >> 6, j = i & 63;
    size_t rb = (rowbase + t) * CONV_DIM;
#ifndef HAVE_ASYNC_LDS
    sB[t * ST + j] = xbcc[rb + D_INNER + j];
#endif
    sU[t * ST + j] = dtsp[(rowbase + t) * NHEADS + h] * xbcc[rb + h * HEADDIM + j];
  }
  if (tid < LC) sa[tid] = dAv[(rowbase + tid) * NHEADS + h];
#ifdef HAVE_ASYNC_LDS
  async_wait0();
#endif
  __syncthreads();
  if (tid == 0) {
    float p = 1.0f;
    for (int t = 0; t < LC; ++t) { p *= sa[t]; scum[t] = p; }
  }
  __syncthreads();
  if (tid < LC) sic[tid] = 1.0f / scum[tid];
  __syncthreads();

  v8f acc[4] = {};
  const int prow = wrow + l15;
  for (int kk = 0; kk < 64; kk += 4) {  // k = s
    const int ka = kk + hl * 2;
    v2f a;
    a.x = sic[ka] * sU[ka * ST + prow];
    a.y = sic[ka + 1] * sU[(ka + 1) * ST + prow];
#pragma unroll
    for (int j = 0; j < 4; ++j) {
      const int col = j * 16 + l15;  // n
      v2f bb;
      bb.x = sB[ka * ST + col];
      bb.y = sB[(ka + 1) * ST + col];
      acc[j] = wmma4(a, bb, acc[j]);
    }
  }
  const float cl = scum[LC - 1];
  const size_t base = ((size_t)bh * NCHUNK + cidx) * (LC * 64);
#pragma unroll
  for (int j = 0; j < 4; ++j)
#pragma unroll
    for (int i = 0; i < 8; ++i) {
      int p = wrow + i + hl * 8, n = j * 16 + l15;
      Sloc[base + p * 64 + n] = cl * acc[j][i];
    }
  if (tid == 0) cumL[bh * NCHUNK + cidx] = cl;
}

// ---- Pass B: prefix over chunks; Sloc[c] := H_c (entering state), in place
__global__ __launch_bounds__(256) void scan_prefix(
    float* __restrict__ Sloc, const float* __restrict__ cumL) {
  const int bh = blockIdx.x;
  const int tid = threadIdx.x;
  float hreg[16];
#pragma unroll
  for (int e = 0; e < 16; ++e) hreg[e] = 0.0f;
  for (int c = 0; c < NCHUNK; ++c) {
    const size_t base = ((size_t)bh * NCHUNK + c) * (LC * 64);
    const float cl = cumL[bh * NCHUNK + c];
#pragma unroll
    for (int e = 0; e < 16; ++e) {
      int i = tid + e * 256;
      float s = Sloc[base + i];
      Sloc[base + i] = hreg[e];           // entering state H_c
      hreg[e] = cl * hreg[e] + s;         // H_{c+1}
    }
  }
}

// ---- Pass C: per-(b,h,chunk) outputs:
//   Y = (mask.decay o C B^T) U  +  (cum_t C) H_c^T  +  D*x
__global__ __launch_bounds__(128) void scan_outputC(
    const float* __restrict__ xbcc, const float* __restrict__ dtsp,
    const float* __restrict__ dAv, const float* __restrict__ Dp,
    const float* __restrict__ Sloc, float* __restrict__ ybuf) {
  extern __shared__ float sm[];
  float* sB = sm;             // LC x ST
  float* sC = sB + LC * ST;   // LC x ST
  float* sU = sC + LC * ST;   // LC x ST
  float* sG = sU + LC * ST;   // LC x ST
  float* sH = sG + LC * ST;   // P  x ST
  float* sa = sH + LC * ST;
  float* scum = sa + LC;
  float* sic  = scum + LC;

  const int bh = blockIdx.x, cidx = blockIdx.y;
  const int b = bh / NHEADS, h = bh % NHEADS;
  const int tid = threadIdx.x;
  const int wave = tid >> 5, lane = tid & 31, hl = lane >> 4, l15 = lane & 15;
  const int wrow = wave * 16;
  const size_t rowbase = (size_t)b * SEQ_L + cidx * LC;
  const size_t hbase = ((size_t)bh * NCHUNK + cidx) * (LC * 64);
  const float Dh = Dp[h];

#ifdef HAVE_ASYNC_LDS
  for (int i = tid; i < LC * 16; i += 128) {
    int t = i >> 4, c4 = (i & 15) * 4;
    size_t rb = (rowbase + t) * CONV_DIM;
    async_cp16(&xbcc[rb + D_INNER + c4], &sB[t * ST + c4]);
    async_cp16(&xbcc[rb + D_INNER + D_STATE + c4], &sC[t * ST + c4]);
  }
#endif
  for (int i = tid; i < LC * 64; i += 128) {
    int t = i >> 6, j = i & 63;
    size_t rb = (rowbase + t) * CONV_DIM;
#ifndef HAVE_ASYNC_LDS
    sB[t * ST + j] = xbcc[rb + D_INNER + j];
    sC[t * ST + j] = xbcc[rb + D_INNER + D_STATE + j];
#endif
    sU[t * ST + j] = dtsp[(rowbase + t) * NHEADS + h] * xbcc[rb + h * HEADDIM + j];
    sH[t * ST + j] = Sloc[hbase + i];   // h[p=t][n=j], entering state
  }
  if (tid < LC) sa[tid] = dAv[(rowbase + tid) * NHEADS + h];
#ifdef HAVE_ASYNC_LDS
  async_wait0();
#endif
  __syncthreads();
  if (tid == 0) {
    float p = 1.0f;
    for (int t = 0; t < LC; ++t) { p *= sa[t]; scum[t] = p; }
  }
  __syncthreads();
  if (tid < LC) sic[tid] = 1.0f / scum[tid];
  __syncthreads();

  // G = C * B^T with causal decay mask
  {
    v8f acc[4] = {};
    for (int kk = 0; kk < 64; kk += 4) {  // k = n
      const int ka = kk + hl * 2;
      v2f a;
      a.x = sC[(wrow + l15) * ST + ka];
      a.y = sC[(wrow + l15) * ST + ka + 1];
#pragma unroll
      for (int j = 0; j < 4; ++j) {
        const int col = j * 16 + l15;  // s
        v2f bb;
        bb.x = sB[col * ST + ka];
        bb.y = sB[col * ST + ka + 1];
        acc[j] = wmma4(a, bb, acc[j]);
      }
    }
#pragma unroll
    for (int j = 0; j < 4; ++j)
#pragma unroll
      for (int i = 0; i < 8; ++i) {
        int t = wrow + i + hl * 8, s = j * 16 + l15;
        sG[t * ST + s] = (s <= t) ? acc[j][i] * scum[t] * sic[s] : 0.0f;
      }
  }
  __syncthreads();

  // Y = G*U + (cum*C)*H^T ; store with D*x
  {
    v8f acc[4] = {};
    for (int kk = 0; kk < 64; kk += 4) {  // intra, k = s
      const int ka = kk + hl * 2;
      v2f a;
      a.x = sG[(wrow + l15) * ST + ka];
      a.y = sG[(wrow + l15) * ST + ka + 1];
#pragma unroll
      for (int j = 0; j < 4; ++j) {
        const int col = j * 16 + l15;  // p
        v2f bb;
        bb.x = sU[ka * ST + col];
        bb.y = sU[(ka + 1) * ST + col];
        acc[j] = wmma4(a, bb, acc[j]);
      }
    }
    const int trow = wrow + l15;
    const float ct = scum[trow];
    for (int kk = 0; kk < 64; kk += 4) {  // inter, k = n
      const int ka = kk + hl * 2;
      v2f a;
      a.x = ct * sC[trow * ST + ka];
      a.y = ct * sC[trow * ST + ka + 1];
#pragma unroll
      for (int j = 0; j < 4; ++j) {
        const int col = j * 16 + l15;  // p
        v2f bb;
        bb.x = sH[col * ST + ka];      // h[p][n]
        bb.y = sH[col * ST + ka + 1];
        acc[j] = wmma4(a, bb, acc[j]);
      }
    }
#pragma unroll
    for (int j = 0; j < 4; ++j)
#pragma unroll
      for (int i = 0; i < 8; ++i) {
        int t = wrow + i + hl * 8, p = j * 16 + l15;
        size_t idx = rowbase + t;
        float xv = xbcc[idx * CONV_DIM + h * HEADDIM + p];
        ybuf[idx * D_INNER + h * HEADDIM + p] = acc[j][i] + Dh * xv;
      }
  }
}

// =====================================================================
// Kernel: y = ybuf * silu(z); RMS-norm over D_INNER; * norm_w
// =====================================================================
__global__ __launch_bounds__(256) void gate_rmsnorm(
    const float* __restrict__ zx, const float* __restrict__ ybuf,
    const float* __restrict__ norm_w, float* __restrict__ ypre) {
  __shared__ float red[8];
  __shared__ float rinv;
  const int row = blockIdx.x;
  const int tid = threadIdx.x;
  const int wave = tid >> 5, lane = tid & 31;
  const float* z = zx + (size_t)row * D_IN_PROJ;
  const float* y = ybuf + (size_t)row * D_INNER;

  float local[6];
  float ss = 0.0f;
#pragma unroll
  for (int e = 0; e < 6; ++e) {
    int i = tid + e * 256;
    float zi = z[i];
    float yi = y[i] * (zi * sigmoidf_(zi));
    local[e] = yi;
    ss += yi * yi;
  }
#pragma unroll
  for (int m = 16; m >= 1; m >>= 1) ss += __shfl_xor(ss, m, 32);
  if (lane == 0) red[wave] = ss;
  __syncthreads();
  if (tid == 0) {
    float t = 0.0f;
    for (int w = 0; w < 8; ++w) t += red[w];
    rinv = rsqrtf(t / (float)D_INNER + EPSV);
  }
  __syncthreads();
  const float r = rinv;
#pragma unroll
  for (int e = 0; e < 6; ++e) {
    int i = tid + e * 256;
    ypre[(size_t)row * D_INNER + i] = local[e] * r * norm_w[i];
  }
}

// =====================================================================
// launch
// =====================================================================
extern "C" void kernel_launch(void* const* d_in, const int* in_sizes, int n_in,
                              void* d_out, int out_size, void* d_ws,
                              size_t ws_size, hipStream_t stream) {
  const float* x       = (const float*)d_in[0];
  const float* W_in    = (const float*)d_in[1];
  const float* conv_w  = (const float*)d_in[2];
  const float* conv_b  = (const float*)d_in[3];
  const float* dt_bias = (const float*)d_in[4];
  const float* A_log   = (const float*)d_in[5];
  const float* Dp      = (const float*)d_in[6];
  const float* norm_w  = (const float*)d_in[7];
  const float* W_out   = (const float*)d_in[8];
  float* out = (float*)d_out;

  float* ws = (float*)d_ws;
  float* zx   = ws;                                   // 8192*3224
  float* xbcc = zx + (size_t)ROWS * D_IN_PROJ;        // 8192*1664
  float* dtsp = xbcc + (size_t)ROWS * CONV_DIM;       // 8192*24
  float* dAv  = dtsp + (size_t)ROWS * NHEADS;         // 8192*24
  float* ybuf = dAv + (size_t)ROWS * NHEADS;          // 8192*1536
  float* Sloc = ybuf + (size_t)ROWS * D_INNER;        // 48*64*4096
  float* cumL = Sloc + (size_t)SEQ_B * NHEADS * NCHUNK * 64 * 64;  // 3072
  float* ypre = xbcc;  // alias: xBC dead after scan pass C

  // 1) in-projection
  gemm_wmma_f32<<<dim3((D_IN_PROJ + 127) / 128, ROWS / 64), 256, 0, stream>>>(
      x, W_in, zx, ROWS, D_IN_PROJ, D_MODEL);

  // 2) dt softplus + dA
  {
    int total = ROWS * NHEADS;
    prep_dt<<<(total + 255) / 256, 256, 0, stream>>>(zx, dt_bias, A_log, dtsp,
                                                     dAv, total);
  }

  // 3) conv + silu
  {
    size_t total = (size_t)ROWS * CONV_DIM;
    conv_silu<<<(unsigned)((total + 255) / 256), 256, 0, stream>>>(zx, conv_w,
                                                                   conv_b, xbcc);
  }

  // 4) SSD scan: A (parallel WMMA) -> B (tiny prefix) -> C (parallel WMMA)
  scan_stateA<<<dim3(SEQ_B * NHEADS, NCHUNK), 128, 0, stream>>>(xbcc, dtsp, dAv,
                                                                Sloc, cumL);
  scan_prefix<<<SEQ_B * NHEADS, 256, 0, stream>>>(Sloc, cumL);
  {
    size_t shmem = (size_t)(5 * LC * ST + 3 * LC) * sizeof(float);  // ~88 KB
    scan_outputC<<<dim3(SEQ_B * NHEADS, NCHUNK), 128, shmem, stream>>>(
        xbcc, dtsp, dAv, Dp, Sloc, ybuf);
  }

  // 5) silu gate + RMS norm
  gate_rmsnorm<<<ROWS, 256, 0, stream>>>(zx, ybuf, norm_w, ypre);

  // 6) out projection
  gemm_wmma_f32<<<dim3((D_MODEL + 127) / 128, ROWS / 64), 256, 0, stream>>>(
      ypre, W_out, out, ROWS, D_MODEL, D_INNER);
}